// GraphSAGE_51496657879701
// MI455X (gfx1250) — compile-verified
//
#include <hip/hip_runtime.h>

typedef __attribute__((ext_vector_type(2))) float v2f;
typedef __attribute__((ext_vector_type(8))) float v8f;

#define N_NODES 100000
#define N_EDGES 640000
#define IN_DIM  128
#define HID_DIM 64
#define OUT_DIM 32

// ---------------------------------------------------------------------------
// Scatter-add of source-node features onto destination nodes.
// One wave (32 lanes) per edge; each lane handles DIM/32 consecutive floats
// (coalesced 512B / 256B burst per edge). Counts accumulated by lane 0.
// unsafeAtomicAdd lowers to global_atomic_add_f32 on gfx1250.
// ---------------------------------------------------------------------------
template <int DIM, bool COUNT>
__global__ void sage_scatter(const float* __restrict__ feat,
                             const int* __restrict__ src,
                             const int* __restrict__ dst,
                             float* __restrict__ agg,
                             float* __restrict__ cnt,
                             int n_edges)
{
    const int wave = (int)((blockIdx.x * (unsigned)blockDim.x + threadIdx.x) >> 5);
    const int lane = threadIdx.x & 31;
    if (wave >= n_edges) return;

    const int s = src[wave];
    const int d = dst[wave];

    constexpr int PER = DIM / 32;             // 4 (dim128) or 2 (dim64)
    const float* fs = feat + (size_t)s * DIM + lane * PER;
    float*       ad = agg  + (size_t)d * DIM + lane * PER;

#pragma unroll
    for (int i = 0; i < PER; ++i)
        unsafeAtomicAdd(ad + i, fs[i]);

    if (COUNT && lane == 0)
        unsafeAtomicAdd(cnt + d, 1.0f);
}

// ---------------------------------------------------------------------------
// agg[i] /= max(cnt[node(i)], 1)   (mean aggregation finalize)
// ---------------------------------------------------------------------------
template <int DIM>
__global__ void sage_mean(float* __restrict__ agg,
                          const float* __restrict__ cnt)
{
    const size_t i = blockIdx.x * (size_t)blockDim.x + threadIdx.x;
    if (i >= (size_t)N_NODES * DIM) return;
    const float c = cnt[i / DIM];
    agg[i] = agg[i] / fmaxf(c, 1.0f);
}

// ---------------------------------------------------------------------------
// out[m, n] = act( A1[m,:]·Wl[n,:] + bias[n] + A2[m,:]·Wr[n,:] )
// A1 = mean-aggregated features, A2 = root features; W* are [NOUT, K] row-major
// (PyTorch layout), so B = W.T tile loads are contiguous float2 per lane.
//
// One wave per 16x16 output tile, K accumulated 4 at a time with
// V_WMMA_F32_16X16X4_F32 (full fp32).
//
// Fragment layouts (CDNA5 ISA 7.12.2, 32-bit):
//   A 16x4 : lane L -> row L%16, VGPR{0,1} hold K = 2*(L/16) + {0,1}
//   B 4x16 : lane L -> col L%16, VGPR{0,1} hold K = 2*(L/16) + {0,1}
//   C/D    : lane L -> col L%16, VGPR v   -> row v + 8*(L/16)
// ---------------------------------------------------------------------------
template <int K, int NOUT, bool RELU>
__global__ __launch_bounds__(128) void sage_gemm(
    const float* __restrict__ A1, const float* __restrict__ Wl,
    const float* __restrict__ bias,
    const float* __restrict__ A2, const float* __restrict__ Wr,
    float* __restrict__ out)
{
    const int warp = threadIdx.x >> 5;        // col tile within block
    const int lane = threadIdx.x & 31;
    const int m0   = blockIdx.x * 16;
    const int n0   = warp * 16;

    const int row   = m0 + (lane & 15);
    const int col   = n0 + (lane & 15);
    const int kpair = (lane >> 4) * 2;        // 0 or 2

    // Accumulator seeded with the bias (column-indexed -> same in all 8 VGPRs)
    v8f acc;
    const float bv = bias[col];
#pragma unroll
    for (int v = 0; v < 8; ++v) acc[v] = bv;

    const float* a1p = A1 + (size_t)row * K + kpair;
    const float* a2p = A2 + (size_t)row * K + kpair;
    const float* w1p = Wl + (size_t)col * K + kpair;
    const float* w2p = Wr + (size_t)col * K + kpair;

#pragma unroll 4
    for (int k = 0; k < K; k += 4) {
        const v2f a1 = *(const v2f*)(a1p + k);
        const v2f b1 = *(const v2f*)(w1p + k);
        acc = __builtin_amdgcn_wmma_f32_16x16x4_f32(
            false, a1, false, b1, (short)0, acc, false, false);

        const v2f a2 = *(const v2f*)(a2p + k);
        const v2f b2 = *(const v2f*)(w2p + k);
        acc = __builtin_amdgcn_wmma_f32_16x16x4_f32(
            false, a2, false, b2, (short)0, acc, false, false);
    }

    const int rbase = (lane >> 4) * 8;        // C/D row layout
#pragma unroll
    for (int v = 0; v < 8; ++v) {
        float val = acc[v];
        if (RELU) val = fmaxf(val, 0.0f);
        out[(size_t)(m0 + rbase + v) * NOUT + col] = val;
    }
}

// ---------------------------------------------------------------------------
// Host-side launch (graph-capture-safe: only kernels + hipMemsetAsync).
// Inputs (setup order): x, edge_index, W1l, b1, W1r, W2l, b2, W2r
// Workspace layout: agg [N*128] (reused as layer-2 agg N*64) | cnt [N] | h [N*64]
// ---------------------------------------------------------------------------
extern "C" void kernel_launch(void* const* d_in, const int* in_sizes, int n_in,
                              void* d_out, int out_size, void* d_ws, size_t ws_size,
                              hipStream_t stream)
{
    const float* x   = (const float*)d_in[0];
    const int*   ei  = (const int*)d_in[1];
    const float* W1l = (const float*)d_in[2];
    const float* b1  = (const float*)d_in[3];
    const float* W1r = (const float*)d_in[4];
    const float* W2l = (const float*)d_in[5];
    const float* b2  = (const float*)d_in[6];
    const float* W2r = (const float*)d_in[7];
    float* out = (float*)d_out;

    const int* src = ei;              // edge_index[0]
    const int* dst = ei + N_EDGES;    // edge_index[1]

    float* agg = (float*)d_ws;                           // N*128 floats
    float* cnt = agg + (size_t)N_NODES * IN_DIM;         // N floats
    float* h   = cnt + N_NODES;                          // N*64 floats

    const int edge_blocks = (N_EDGES + 7) / 8;           // 8 waves/block

    // ---- Layer 1 ----
    hipMemsetAsync(agg, 0, (size_t)N_NODES * IN_DIM * sizeof(float), stream);
    hipMemsetAsync(cnt, 0, (size_t)N_NODES * sizeof(float), stream);

    sage_scatter<IN_DIM, true><<<edge_blocks, 256, 0, stream>>>(
        x, src, dst, agg, cnt, N_EDGES);

    {
        const size_t tot = (size_t)N_NODES * IN_DIM;
        sage_mean<IN_DIM><<<(unsigned)((tot + 255) / 256), 256, 0, stream>>>(agg, cnt);
    }

    sage_gemm<IN_DIM, HID_DIM, true><<<N_NODES / 16, 128, 0, stream>>>(
        agg, W1l, b1, x, W1r, h);

    // ---- Layer 2 (reuse agg buffer, first N*64 floats) ----
    hipMemsetAsync(agg, 0, (size_t)N_NODES * HID_DIM * sizeof(float), stream);

    sage_scatter<HID_DIM, false><<<edge_blocks, 256, 0, stream>>>(
        h, src, dst, agg, cnt, N_EDGES);

    {
        const size_t tot = (size_t)N_NODES * HID_DIM;
        sage_mean<HID_DIM><<<(unsigned)((tot + 255) / 256), 256, 0, stream>>>(agg, cnt);
    }

    sage_gemm<HID_DIM, OUT_DIM, false><<<N_NODES / 16, 64, 0, stream>>>(
        agg, W2l, b2, h, W2r, out);
}